// KVCacheAttention_53970559042324
// MI455X (gfx1250) — compile-verified
//
#include <hip/hip_runtime.h>
#include <hip/hip_bf16.h>

typedef __attribute__((ext_vector_type(2))) float v2f;
typedef __attribute__((ext_vector_type(4))) float v4f;
typedef __attribute__((ext_vector_type(8))) float v8f;

#define KB    16
#define KH    16
#define KD    2048
#define KHD   128
#define KPAST 4096
#define KTOT  4097

// ---------------------------------------------------------------------------
// GEMM: Y[mat] = X(16 x 2048) * W[mat]^T   (nn.Linear), via V_WMMA_F32_16X16X4_F32.
// One wave owns one 16x16 output tile; K-loop steps by 4.
//
// fp32 WMMA operand layouts (ISA 7.12.2, 32-bit A 16x4 / B 4x16 / C-D 16x16):
//   A: lane L holds rows M=L&15, a.x=K=(L>>4)*2, a.y=K+1
//   B: lane L holds col  N=L&15, b.x=K=(L>>4)*2, b.y=K+1   (mirror of A)
//   C/D: VGPR v, lane L -> (M = v + 8*(L>>4), N = L&15)
// ---------------------------------------------------------------------------
__global__ void gemm16_wmma_kernel(const float* __restrict__ X,
                                   const float* __restrict__ W0,
                                   const float* __restrict__ W1,
                                   const float* __restrict__ W2,
                                   float* __restrict__ Y,
                                   int nmat)
{
    const int tilesPerMat = KD / 16;                       // 128
    int gw   = (int)((blockIdx.x * blockDim.x + threadIdx.x) >> 5);
    int lane = (int)(threadIdx.x & 31);
    if (gw >= nmat * tilesPerMat) return;
    int mat  = gw / tilesPerMat;
    int tile = gw - mat * tilesPerMat;
    const float* W = (mat == 0) ? W0 : ((mat == 1) ? W1 : W2);
    float* Ym = Y + (size_t)mat * (KB * KD);

    const int j0   = tile * 16;
    const int mrow = lane & 15;
    const int kk   = (lane >> 4) << 1;                     // 0 or 2
    const float* xp = X + (size_t)mrow * KD + kk;          // A: X row = batch row
    const float* wp = W + (size_t)(j0 + mrow) * KD + kk;   // B: W^T col n = W row j0+n

    v8f acc = {};
    for (int k = 0; k < KD; k += 4) {
        v2f a = *(const v2f*)(xp + k);
        v2f b = *(const v2f*)(wp + k);
        acc = __builtin_amdgcn_wmma_f32_16x16x4_f32(false, a, false, b,
                                                    (short)0, acc, false, false);
    }

    const int rbase = (lane >> 4) << 3;
    const int col   = j0 + (lane & 15);
#pragma unroll
    for (int v = 0; v < 8; ++v)
        Ym[(size_t)(rbase + v) * KD + col] = acc[v];
}

// ---------------------------------------------------------------------------
// Scatter freshly-projected k/v rows into present_{k,v}[:, :, 4096, :].
// ---------------------------------------------------------------------------
__global__ void scatter_newkv_kernel(const float* __restrict__ kw,
                                     const float* __restrict__ vw,
                                     float* __restrict__ pk,
                                     float* __restrict__ pv)
{
    int i = blockIdx.x * blockDim.x + threadIdx.x;
    if (i >= KB * KD) return;
    int b   = i >> 11;
    int col = i & (KD - 1);
    int h   = col >> 7;
    int e   = col & (KHD - 1);
    size_t off = (((size_t)b * KH + h) * KTOT + KPAST) * KHD + e;
    pk[off] = kw[i];
    pv[off] = vw[i];
}

// ---------------------------------------------------------------------------
// Fused: single pass over past_k/past_v per (b,h).
//   - streams each K/V row once (nontemporal b128 loads)
//   - writes the present_{k,v} concat copy (nontemporal stores, never re-read)
//   - flash-style online softmax: per-wave (m, l, acc[128]), acc distributed
//     4 floats/lane; cross-lane dot reduction via wave32 shfl_xor;
//     final combine of 16 waves through LDS.
// ---------------------------------------------------------------------------
__global__ __launch_bounds__(512)
void attn_fused_kernel(const float* __restrict__ qws,
                       const float* __restrict__ kws,
                       const float* __restrict__ vws,
                       const float* __restrict__ past_k,
                       const float* __restrict__ past_v,
                       float* __restrict__ pk,
                       float* __restrict__ pv,
                       float* __restrict__ aout)
{
    constexpr int NW = 16;                                  // waves per block
    int bh   = blockIdx.x;
    int b    = bh >> 4, h = bh & 15;
    int lane = threadIdx.x & 31;
    int wave = threadIdx.x >> 5;
    const float scale = 0.08838834764831845f;               // 1/sqrt(128)

    v4f qv = *(const v4f*)(qws + (size_t)b * KD + h * KHD + 4 * lane);

    size_t base  = ((size_t)b * KH + h) * (size_t)KPAST * KHD;
    size_t pbase = ((size_t)b * KH + h) * (size_t)KTOT  * KHD;
    const float* Kp   = past_k + base;
    const float* Vp   = past_v + base;
    const float* knew = kws + (size_t)b * KD + h * KHD;
    const float* vnew = vws + (size_t)b * KD + h * KHD;

    float m = -__builtin_inff();
    float l = 0.f;
    v4f acc = {};

    for (int t = wave; t < KTOT; t += NW) {
        v4f kr, vr;
        if (t < KPAST) {
            kr = __builtin_nontemporal_load((const v4f*)(Kp + (size_t)t * KHD) + lane);
            vr = __builtin_nontemporal_load((const v4f*)(Vp + (size_t)t * KHD) + lane);
            __builtin_nontemporal_store(kr, (v4f*)(pk + pbase + (size_t)t * KHD) + lane);
            __builtin_nontemporal_store(vr, (v4f*)(pv + pbase + (size_t)t * KHD) + lane);
        } else {                                            // t == 4096: new row
            kr = *((const v4f*)knew + lane);
            vr = *((const v4f*)vnew + lane);
        }
        // dot(q, k_row): 4 elems/lane, then wave32 butterfly reduce
        float part = qv.x * kr.x + qv.y * kr.y + qv.z * kr.z + qv.w * kr.w;
        part += __shfl_xor(part, 1, 32);
        part += __shfl_xor(part, 2, 32);
        part += __shfl_xor(part, 4, 32);
        part += __shfl_xor(part, 8, 32);
        part += __shfl_xor(part, 16, 32);
        float s  = part * scale;
        float mn = fmaxf(m, s);
        float corr = __expf(m - mn);
        float p    = __expf(s - mn);
        acc = acc * corr + vr * p;
        l   = l * corr + p;
        m   = mn;
    }

    __shared__ float sm[NW], sl[NW];
    __shared__ v4f   sacc[NW][32];
    sacc[wave][lane] = acc;
    if (lane == 0) { sm[wave] = m; sl[wave] = l; }
    __syncthreads();

    if (wave == 0) {
        float M = -__builtin_inff();
#pragma unroll
        for (int w = 0; w < NW; ++w) M = fmaxf(M, sm[w]);
        float L = 0.f;
        v4f o = {};
#pragma unroll
        for (int w = 0; w < NW; ++w) {
            float f = __expf(sm[w] - M);
            L += f * sl[w];
            o = o + sacc[w][lane] * f;
        }
        float inv = 1.0f / L;
        *(v4f*)(aout + (size_t)b * KD + h * KHD + 4 * lane) = o * inv;
    }
}

// ---------------------------------------------------------------------------
// inputs: x, past_k, past_v, Wq, Wk, Wv, Wo (all fp32)
// d_out:  output(16x2048) | present_k(16x16x4097x128) | present_v(same)
// ---------------------------------------------------------------------------
extern "C" void kernel_launch(void* const* d_in, const int* in_sizes, int n_in,
                              void* d_out, int out_size, void* d_ws, size_t ws_size,
                              hipStream_t stream)
{
    const float* x      = (const float*)d_in[0];
    const float* past_k = (const float*)d_in[1];
    const float* past_v = (const float*)d_in[2];
    const float* Wq     = (const float*)d_in[3];
    const float* Wk     = (const float*)d_in[4];
    const float* Wv     = (const float*)d_in[5];
    const float* Wo     = (const float*)d_in[6];

    float* out = (float*)d_out;
    float* pk  = out + (size_t)KB * KD;
    float* pv  = pk + (size_t)KB * KH * KTOT * KHD;

    float* qws = (float*)d_ws;                 // [16][2048]
    float* kws = qws + KB * KD;
    float* vws = kws + KB * KD;
    float* aws = vws + KB * KD;                // attention output [16][2048]

    // Q,K,V projections: 3*128 tiles = 384 waves = 48 blocks of 8 waves
    gemm16_wmma_kernel<<<48, 256, 0, stream>>>(x, Wq, Wk, Wv, qws, 3);
    // new k/v -> present row 4096
    scatter_newkv_kernel<<<(KB * KD + 255) / 256, 256, 0, stream>>>(kws, vws, pk, pv);
    // fused cache-copy + attention (one pass over past_k/past_v)
    attn_fused_kernel<<<KB * KH, 512, 0, stream>>>(qws, kws, vws, past_k, past_v,
                                                   pk, pv, aws);
    // output projection: 128 tiles = 16 blocks
    gemm16_wmma_kernel<<<16, 256, 0, stream>>>(aws, Wo, Wo, Wo, out, 1);
}